// ScatteringModule_4466765988507
// MI455X (gfx1250) — compile-verified
//
#include <hip/hip_runtime.h>

typedef __attribute__((ext_vector_type(2))) float v2f;
typedef __attribute__((ext_vector_type(8))) float v8f;

#define BB 64
#define CC 128
#define ORI 6
#define HH 28
#define WW 28
#define NBLK 4
#define BS 32
#define HW (HH * WW)                    /* 784 */
#define XH_C_STRIDE (ORI * HW * 2)      /* 9408 floats */
#define XH_B_STRIDE (CC * XH_C_STRIDE)  /* 1204224 floats */
#define XH_O_STRIDE (HW * 2)            /* 1568 floats */
#define XL_SIZE (BB * CC * HW)          /* 6422528 floats */
#define TILES_PER_BO (HW / 16)          /* 49 */
#define TOTAL_WAVES (BB * ORI * TILES_PER_BO * NBLK) /* 75264 */
#define WAVES_PER_WG 8
#define NBLOCKS (TOTAL_WAVES / WAVES_PER_WG)         /* 9408 */

static __device__ __forceinline__ v8f wmma4(v2f a, v2f b, v8f c) {
  // D = A(16x4 f32) * B(4x16 f32) + C(16x16 f32)
  return __builtin_amdgcn_wmma_f32_16x16x4_f32(
      /*neg_a=*/false, a, /*neg_b=*/false, b,
      /*c_mod=*/(short)0, c, /*reuse_a=*/false, /*reuse_b=*/false);
}

// ---------------------------------------------------------------------------
// Lowpass: xl_out = xl * w_ll   (w_ll broadcast over batch). Memory-bound.
// ---------------------------------------------------------------------------
__global__ __launch_bounds__(256) void scat_lowpass_kernel(
    const float* __restrict__ xl, const float* __restrict__ wll,
    float* __restrict__ out) {
  const int i = blockIdx.x * blockDim.x + threadIdx.x;  // float4 index
  const int N4 = XL_SIZE / 4;
  if (i >= N4) return;
  const int WN4 = (CC * HW) / 4;  // 25088
  const float4 x = ((const float4*)xl)[i];
  const float4 w = ((const float4*)wll)[i % WN4];
  float4 r;
  r.x = x.x * w.x; r.y = x.y * w.y; r.z = x.z * w.z; r.w = x.w * w.w;
  ((float4*)out)[i] = r;
}

// ---------------------------------------------------------------------------
// Highpass: complex block-diagonal 2-layer MLP over channels, WMMA fp32.
// One wave handles a tile of 16 spatial positions x one 32-channel block.
// ---------------------------------------------------------------------------
__global__ __launch_bounds__(256) void scat_xh_kernel(
    const float* __restrict__ xh, const float* __restrict__ w1,
    const float* __restrict__ w2, const float* __restrict__ b1,
    const float* __restrict__ b2, float* __restrict__ out_xh) {
  // Per-wave LDS scratch: 1088 dwords. Used first as two 16x32 tiles with
  // row stride 34 (r1 @ 0, i1 @ 544), then reused as one 16x32 float2 tile
  // with row stride 68 for the output transpose.
  __shared__ __align__(16) float lds[WAVES_PER_WG][1088];

  const int wid = threadIdx.x >> 5;
  const int lane = threadIdx.x & 31;
  const int lane16 = lane & 15;
  const int lhalf = lane >> 4;

  const int gw = blockIdx.x * WAVES_PER_WG + wid;
  const int nb = gw & (NBLK - 1);
  const int t = gw >> 2;
  const int b = t / (ORI * TILES_PER_BO);
  const int rem = t % (ORI * TILES_PER_BO);
  const int ori = rem / TILES_PER_BO;
  const int qbase = (rem % TILES_PER_BO) * 16;  // 16 consecutive (h,w) slots

  // element (pos, ch): base + ch*XH_C_STRIDE + pos*2  (+0 re, +1 im)
  const size_t base = (size_t)b * XH_B_STRIDE + (size_t)ori * XH_O_STRIDE +
                      (size_t)(nb * BS) * XH_C_STRIDE + (size_t)qbase * 2;
  const float* xin = xh + base;

  // ---- Load A-fragments of Xr / Xi in WMMA A-layout -----------------------
  // A 16x4 f32: lanes0-15 hold M=lane, comps = K{0,1}; lanes16-31 K{2,3}.
  v2f ar[8], ai[8];
#pragma unroll
  for (int kc = 0; kc < 8; ++kc) {
    const int ch = kc * 4 + 2 * lhalf;
    const float2 u0 = *(const float2*)(xin + (size_t)ch * XH_C_STRIDE + lane16 * 2);
    const float2 u1 = *(const float2*)(xin + (size_t)(ch + 1) * XH_C_STRIDE + lane16 * 2);
    ar[kc] = (v2f){u0.x, u1.x};
    ai[kc] = (v2f){u0.y, u1.y};
  }

  const float* w1r = w1 + nb * (BS * BS);
  const float* w1i = w1 + NBLK * BS * BS + nb * (BS * BS);
  float* sr1 = &lds[wid][0];
  float* si1 = &lds[wid][544];

  // ---- Layer 1: R1 = relu(Xr*W1r - Xi*W1i + b1r), I1 = relu(Xr*W1i + Xi*W1r + b1i)
#pragma unroll
  for (int nh = 0; nh < 2; ++nh) {
    v8f aR = {0.f, 0.f, 0.f, 0.f, 0.f, 0.f, 0.f, 0.f};
    v8f aI = {0.f, 0.f, 0.f, 0.f, 0.f, 0.f, 0.f, 0.f};
#pragma unroll
    for (int kc = 0; kc < 8; ++kc) {
      const int kb = kc * 4 + 2 * lhalf;        // B-layout mirrors A K slots
      const int n = nh * 16 + lane16;
      const v2f br = (v2f){w1r[kb * BS + n], w1r[(kb + 1) * BS + n]};
      const v2f bi = (v2f){w1i[kb * BS + n], w1i[(kb + 1) * BS + n]};
      const v2f nbi = -bi;                      // f32 WMMA NEG can't negate B
      aR = wmma4(ar[kc], br, aR);
      aR = wmma4(ai[kc], nbi, aR);
      aI = wmma4(ar[kc], bi, aI);
      aI = wmma4(ai[kc], br, aI);
    }
    const float br1 = b1[nb * BS + nh * 16 + lane16];
    const float bi1 = b1[NBLK * BS + nb * BS + nh * 16 + lane16];
#pragma unroll
    for (int v = 0; v < 8; ++v) {
      const int M = v + 8 * lhalf;              // D-layout row
      float r = aR[v] + br1; r = r > 0.f ? r : 0.f;
      float im = aI[v] + bi1; im = im > 0.f ? im : 0.f;
      sr1[M * 34 + nh * 16 + lane16] = r;
      si1[M * 34 + nh * 16 + lane16] = im;
    }
  }
  __syncthreads();

  // ---- Re-gather layer-1 outputs in A-layout ------------------------------
  v2f cr[8], ci[8];
#pragma unroll
  for (int kc = 0; kc < 8; ++kc) {
    const int kk = kc * 4 + 2 * lhalf;          // even -> 8B aligned
    cr[kc] = *(const v2f*)&sr1[lane16 * 34 + kk];
    ci[kc] = *(const v2f*)&si1[lane16 * 34 + kk];
  }
  __syncthreads();  // all frag reads done before LDS is reused below

  // ---- Layer 2 (linear) + output transpose --------------------------------
  const float* w2r = w2 + nb * (BS * BS);
  const float* w2i = w2 + NBLK * BS * BS + nb * (BS * BS);
  float* sf = &lds[wid][0];  // float2 tile, row stride 68 dwords
#pragma unroll
  for (int nh = 0; nh < 2; ++nh) {
    v8f aR = {0.f, 0.f, 0.f, 0.f, 0.f, 0.f, 0.f, 0.f};
    v8f aI = {0.f, 0.f, 0.f, 0.f, 0.f, 0.f, 0.f, 0.f};
#pragma unroll
    for (int kc = 0; kc < 8; ++kc) {
      const int kb = kc * 4 + 2 * lhalf;
      const int n = nh * 16 + lane16;
      const v2f br = (v2f){w2r[kb * BS + n], w2r[(kb + 1) * BS + n]};
      const v2f bi = (v2f){w2i[kb * BS + n], w2i[(kb + 1) * BS + n]};
      const v2f nbi = -bi;
      aR = wmma4(cr[kc], br, aR);
      aR = wmma4(ci[kc], nbi, aR);
      aI = wmma4(cr[kc], bi, aI);
      aI = wmma4(ci[kc], br, aI);
    }
    const float br2 = b2[nb * BS + nh * 16 + lane16];
    const float bi2 = b2[NBLK * BS + nb * BS + nh * 16 + lane16];
#pragma unroll
    for (int v = 0; v < 8; ++v) {
      const int M = v + 8 * lhalf;
      const int ch = nh * 16 + lane16;
      *(float2*)&sf[M * 68 + ch * 2] = make_float2(aR[v] + br2, aI[v] + bi2);
    }
  }
  __syncthreads();

  // Write out: per channel, 16 positions * float2 = 128B contiguous.
  float* xo = out_xh + base;
#pragma unroll
  for (int it = 0; it < 16; ++it) {
    const int ch = it * 2 + lhalf;
    const float2 val = *(const float2*)&sf[lane16 * 68 + ch * 2];
    *(float2*)(xo + (size_t)ch * XH_C_STRIDE + lane16 * 2) = val;
  }
}

// ---------------------------------------------------------------------------
extern "C" void kernel_launch(void* const* d_in, const int* in_sizes, int n_in,
                              void* d_out, int out_size, void* d_ws,
                              size_t ws_size, hipStream_t stream) {
  const float* xl  = (const float*)d_in[0];
  const float* xh  = (const float*)d_in[1];
  const float* wll = (const float*)d_in[2];
  const float* w1  = (const float*)d_in[3];
  const float* w2  = (const float*)d_in[4];
  const float* b1  = (const float*)d_in[5];
  const float* b2  = (const float*)d_in[6];
  float* out = (float*)d_out;

  scat_lowpass_kernel<<<(XL_SIZE / 4 + 255) / 256, 256, 0, stream>>>(xl, wll, out);
  scat_xh_kernel<<<NBLOCKS, 256, 0, stream>>>(xh, w1, w2, b1, b2, out + XL_SIZE);
}